// QuantumBlock_64312840290525
// MI455X (gfx1250) — compile-verified
//
#include <hip/hip_runtime.h>

typedef __attribute__((ext_vector_type(2))) float v2f;
typedef __attribute__((ext_vector_type(8))) float v8f;

#define NQ      14
#define NSTATE  16384      // 2^14 amplitudes
#define TPB     256        // 8 wave32s per workgroup
#define NBLK    4
#define NWAVE   (TPB / 32)

__device__ __forceinline__ int insert_zero(int x, int b) {
    return ((x >> b) << (b + 1)) | (x & ((1 << b) - 1));
}

// Apply real 16x16 unitary U (row-major in LDS) to the 4-bit index field at
// bit position g of every amplitude:  S'[m][k'] = sum_k S[m][k] * U[k'][k].
// State viewed as 1024 rows x 16 cols; B-operand = U^T, K=16 via 4 chained
// V_WMMA_F32_16X16X4_F32. Uniform trip count -> EXEC stays all-ones.
__device__ __forceinline__ void apply_u16_wmma(float* __restrict__ Sr,
                                               float* __restrict__ Si,
                                               const float* __restrict__ U,
                                               int g, int wid, int h, int ln)
{
    v2f bf[4];
    #pragma unroll
    for (int ks = 0; ks < 4; ++ks) {
        int k0 = 4 * ks + 2 * h;            // B frag: VGPR0<->K=k0, VGPR1<->K=k0+1, N=ln
        bf[ks].x = U[ln * 16 + k0];         // B[k][n] = U[n][k]
        bf[ks].y = U[ln * 16 + k0 + 1];
    }
    const int gm = (1 << g) - 1;
    #pragma unroll 1
    for (int it = 0; it < 64 / NWAVE; ++it) {   // 64 chunks / 8 waves = 8, uniform
        int row0 = (wid + it * NWAVE) * 16;
        int m    = row0 + ln;                       // A frag: M = ln
        int base = ((m >> g) << (g + 4)) | (m & gm);
        v2f ar[4], ai[4];
        #pragma unroll
        for (int ks = 0; ks < 4; ++ks) {
            int k0 = 4 * ks + 2 * h;
            int a0 = base | (k0 << g);
            int a1 = base | ((k0 + 1) << g);
            ar[ks].x = Sr[a0];  ar[ks].y = Sr[a1];
            ai[ks].x = Si[a0];  ai[ks].y = Si[a1];
        }
        v8f cr = {0.f, 0.f, 0.f, 0.f, 0.f, 0.f, 0.f, 0.f};
        v8f ci = {0.f, 0.f, 0.f, 0.f, 0.f, 0.f, 0.f, 0.f};
        #pragma unroll
        for (int ks = 0; ks < 4; ++ks) {
            cr = __builtin_amdgcn_wmma_f32_16x16x4_f32(
                false, ar[ks], false, bf[ks], (short)0, cr, false, false);
            ci = __builtin_amdgcn_wmma_f32_16x16x4_f32(
                false, ai[ks], false, bf[ks], (short)0, ci, false, false);
        }
        #pragma unroll
        for (int j = 0; j < 8; ++j) {
            int md = row0 + 8 * h + j;              // C/D: VGPR j -> M = j+8h, N = ln
            int ad = ((md >> g) << (g + 4)) | (md & gm) | (ln << g);
            Sr[ad] = cr[j];
            Si[ad] = ci[j];
        }
    }
}

__global__ __launch_bounds__(TPB)
void qsim_kernel(const float* __restrict__ xin,   // [256][14]
                 const float* __restrict__ w,     // [4][3][14]
                 float* __restrict__ out)         // [256][14]
{
    extern __shared__ float smem[];
    float* Sr   = smem;               // 16384 real plane
    float* Si   = Sr + NSTATE;        // 16384 imag plane
    float* U    = Si + NSTATE;        // 256   fused 16x16 unitary
    float* acc  = U + 256;            // 16    output accumulators
    float* gc   = acc + 16;           // 16    per-layer cos table
    float* gs   = gc + 16;            // 16    per-layer sin table
    float* colt = gs + 16;            // 32    column (low-4-bit) table
    float* rowt = colt + 32;          // 2048  row (high-10-bit) table
    // total: 32768 + 256 + 16 + 16 + 16 + 32 + 2048 = 35152 floats = 140608 B

    const int tid  = threadIdx.x;
    const int bat  = blockIdx.x;
    const int lane = tid & 31;
    const int wid  = tid >> 5;
    const int h    = (lane >> 4) & 1;
    const int ln   = lane & 15;

    // ---------------- init: product state of RY(x)|0>, factored row x col ----
    if (tid < NQ) {
        float th = 0.5f * xin[bat * NQ + tid];
        gc[tid] = cosf(th);
        gs[tid] = sinf(th);
    }
    __syncthreads();
    for (int idx = tid; idx < 1024; idx += TPB) {
        float p = 1.0f;
        #pragma unroll
        for (int q = 0; q < 10; ++q)              // qubit q -> bit (9-q) of hi field
            p *= ((idx >> (9 - q)) & 1) ? gs[q] : gc[q];
        rowt[idx] = p;
    }
    if (tid < 16) {
        float p = 1.0f;
        #pragma unroll
        for (int q = 10; q < NQ; ++q)             // qubit q -> bit (13-q) of lo field
            p *= ((tid >> (13 - q)) & 1) ? gs[q] : gc[q];
        colt[tid] = p;
    }
    __syncthreads();
    for (int i = tid; i < NSTATE; i += TPB) {
        Sr[i] = rowt[i >> 4] * colt[i & 15];
        Si[i] = 0.0f;
    }
    __syncthreads();

    // ---------------- 4 entangling blocks ----------------
    for (int blk = 0; blk < NBLK; ++blk) {
        const int wy = (blk * 3 + 0) * NQ;
        const int wz = (blk * 3 + 1) * NQ;
        const int wx = (blk * 3 + 2) * NQ;

        // ---- RY layer: 14 qubits in 4 fused 16x16 WMMA passes ----
        if (tid < NQ) {
            float th = 0.5f * w[wy + tid];
            gc[tid] = cosf(th);
            gs[tid] = sinf(th);
        }
        __syncthreads();
        {
            // group j-th factor acts on k-bit (3-j); qubit list per group (-1 = identity)
            constexpr int GQ[4][4] = {{2, 3, 4, 5}, {6, 7, 8, 9}, {10, 11, 12, 13}, {0, 1, -1, -1}};
            constexpr int GG[4]    = {8, 4, 0, 10};
            #pragma unroll
            for (int grp = 0; grp < 4; ++grp) {
                {
                    int cp = tid >> 4, cc = tid & 15;
                    float u = 1.0f;
                    #pragma unroll
                    for (int j = 0; j < 4; ++j) {
                        constexpr int dummy = 0; (void)dummy;
                        int q  = GQ[grp][j];
                        int rb = (cp >> (3 - j)) & 1;
                        int cb = (cc >> (3 - j)) & 1;
                        float f;
                        if (q >= 0)
                            f = (rb == cb) ? gc[q] : ((rb == 0) ? -gs[q] : gs[q]);
                        else
                            f = (rb == cb) ? 1.0f : 0.0f;
                        u *= f;
                    }
                    U[tid] = u;
                }
                __syncthreads();
                apply_u16_wmma(Sr, Si, U, GG[grp], wid, h, ln);
                __syncthreads();
            }
        }

        // ---- CNOT ring (order-dependent, q = 0..13) ----
        for (int q = 0; q < NQ; ++q) {
            int t  = (q + 1) % NQ;
            int bc = 13 - q, bt = 13 - t;
            int b1 = bc < bt ? bc : bt;
            int b2 = bc < bt ? bt : bc;
            for (int p = tid; p < NSTATE / 4; p += TPB) {
                int i  = insert_zero(insert_zero(p, b1), b2);
                int i0 = i | (1 << bc);           // control=1, target=0
                int i1 = i0 | (1 << bt);
                float tr = Sr[i0]; Sr[i0] = Sr[i1]; Sr[i1] = tr;
                float ti = Si[i0]; Si[i0] = Si[i1]; Si[i1] = ti;
            }
            __syncthreads();
        }

        // ---- RZ layer: diagonal; separable phase tables, no trig in hot loop ----
        {
            float thh[NQ];
            #pragma unroll
            for (int q = 0; q < NQ; ++q) thh[q] = 0.5f * w[wz + q];
            for (int idx = tid; idx < 1024; idx += TPB) {
                float ph = 0.0f;
                #pragma unroll
                for (int q = 0; q < 10; ++q)
                    ph += ((idx >> (9 - q)) & 1) ? thh[q] : -thh[q];
                float s, c;
                sincosf(ph, &s, &c);
                rowt[2 * idx]     = c;
                rowt[2 * idx + 1] = s;
            }
            if (tid < 16) {
                float ph = 0.0f;
                #pragma unroll
                for (int q = 10; q < NQ; ++q)
                    ph += ((tid >> (13 - q)) & 1) ? thh[q] : -thh[q];
                float s, c;
                sincosf(ph, &s, &c);
                colt[2 * tid]     = c;
                colt[2 * tid + 1] = s;
            }
            __syncthreads();
            for (int i = tid; i < NSTATE; i += TPB) {
                int hi = i >> 4, lo = i & 15;
                float c1 = rowt[2 * hi], s1 = rowt[2 * hi + 1];
                float c2 = colt[2 * lo], s2 = colt[2 * lo + 1];
                float cp = c1 * c2 - s1 * s2;
                float sp = s1 * c2 + c1 * s2;
                float r = Sr[i], im = Si[i];
                Sr[i] = r * cp - im * sp;
                Si[i] = r * sp + im * cp;
            }
            __syncthreads();
        }

        // ---- CRX ring (order-dependent, q = 0..13) ----
        if (tid < NQ) {
            float th = 0.5f * w[wx + tid];
            gc[tid] = cosf(th);
            gs[tid] = sinf(th);
        }
        __syncthreads();
        for (int q = 0; q < NQ; ++q) {
            int t  = (q + 1) % NQ;
            int bc = 13 - q, bt = 13 - t;
            int b1 = bc < bt ? bc : bt;
            int b2 = bc < bt ? bt : bc;
            float cg = gc[q], sg = gs[q];
            for (int p = tid; p < NSTATE / 4; p += TPB) {
                int i  = insert_zero(insert_zero(p, b1), b2);
                int i0 = i | (1 << bc);
                int i1 = i0 | (1 << bt);
                float a0r = Sr[i0], a0i = Si[i0];
                float a1r = Sr[i1], a1i = Si[i1];
                Sr[i0] = cg * a0r + sg * a1i;
                Si[i0] = cg * a0i - sg * a1r;
                Sr[i1] = cg * a1r + sg * a0i;
                Si[i1] = cg * a1i - sg * a0r;
            }
            __syncthreads();
        }
    }

    // ---------------- Z expectations ----------------
    {
        float a[NQ];
        #pragma unroll
        for (int q = 0; q < NQ; ++q) a[q] = 0.0f;
        for (int i = tid; i < NSTATE; i += TPB) {
            float p = Sr[i] * Sr[i] + Si[i] * Si[i];
            #pragma unroll
            for (int q = 0; q < NQ; ++q)
                a[q] += ((i >> (13 - q)) & 1) ? -p : p;
        }
        if (tid < NQ) acc[tid] = 0.0f;
        __syncthreads();
        #pragma unroll
        for (int q = 0; q < NQ; ++q)
            atomicAdd(&acc[q], a[q]);
        __syncthreads();
        if (tid < NQ)
            out[bat * NQ + tid] = acc[tid];
    }
}

extern "C" void kernel_launch(void* const* d_in, const int* in_sizes, int n_in,
                              void* d_out, int out_size, void* d_ws, size_t ws_size,
                              hipStream_t stream) {
    (void)in_sizes; (void)n_in; (void)d_ws; (void)ws_size; (void)out_size;
    const float* xin = (const float*)d_in[0];
    const float* w   = (const float*)d_in[1];
    float* out = (float*)d_out;

    const size_t shmem = (size_t)(2 * NSTATE + 256 + 16 + 16 + 16 + 32 + 2048) * sizeof(float);
    hipFuncSetAttribute((const void*)qsim_kernel,
                        hipFuncAttributeMaxDynamicSharedMemorySize, (int)shmem);
    qsim_kernel<<<dim3(256), dim3(TPB), shmem, stream>>>(xin, w, out);
}